// RelatEntAtt3D_18734647345346
// MI455X (gfx1250) — compile-verified
//
#include <hip/hip_runtime.h>
#include <math.h>

typedef __attribute__((ext_vector_type(2))) float v2f;
typedef __attribute__((ext_vector_type(8))) float v8f;

#define LRELU_ALPHA 0.2f
#define E_ 256
#define R_ 16
#define F_ 64

// ---------------- workspace layout (floats) ----------------
#define OFF_ADJS 0u          // (R,E,E)  softmax(adj,axis=1), TRANSPOSED [k][j][i]  1048576
#define OFF_ATT  1048576u    // (E,R,F)  attention                                   262144
#define OFF_CMAX 1310720u    // (R,F)    col (over-e) max                              1024
#define OFF_CSUM 1311744u    // (R,F)    col sumexp                                    1024
#define OFF_RMAX 1312768u    // (E,F)    row (over-k) max                             16384
#define OFF_RSUM 1329152u    // (E,F)    row sumexp                                   16384
#define OFF_LOG  1345536u    // (E,R)    alpha logits                                  4096

// K1: adjsT[k][j][i] = softmax_j adj[i,j,k]. block = 256 (j), grid = E*R (i*16+k).
// Transposed store so the WMMA kernel's K-dim (i) reads are contiguous per lane.
__global__ void k_softmax_adj(const float* __restrict__ adj, float* __restrict__ adjsT) {
  __shared__ float red[256];
  const int b = blockIdx.x;
  const int i = b >> 4, k = b & 15;
  const int t = threadIdx.x;                       // j
  const float v = adj[(i * E_ + t) * R_ + k];
  red[t] = v; __syncthreads();
  for (int s = 128; s > 0; s >>= 1) { if (t < s) red[t] = fmaxf(red[t], red[t + s]); __syncthreads(); }
  const float m = red[0]; __syncthreads();
  const float e = __expf(v - m);
  red[t] = e; __syncthreads();
  for (int s = 128; s > 0; s >>= 1) { if (t < s) red[t] += red[t + s]; __syncthreads(); }
  adjsT[(k * E_ + t) * E_ + i] = e / red[0];
}

// K2: WMMA GEMM pair + LeakyReLU-sign fixup -> attention (E,R,F).
// One wave per (k_rel, j-tile, f-tile): grid = 16*16*4 = 1024, block = 32.
__global__ void k_attention_wmma(const float* __restrict__ adjsT,
                                 const float* __restrict__ h,
                                 const float* __restrict__ r,
                                 float* __restrict__ att) {
  const int b  = blockIdx.x;
  const int ft = b & 3;          // f-tile  (4)
  const int jt = (b >> 2) & 15;  // j-tile  (16)
  const int k  = b >> 6;         // relation(16)
  const int lane = threadIdx.x;
  const int half = lane >> 4;    // 0: lanes 0-15, 1: lanes 16-31
  const int l15  = lane & 15;
  const int kb   = half * 2;     // A/B K-sub-offset per ISA 16x4 layout
  const int j_a  = jt * 16 + l15;  // A-matrix M index (j)
  const int f_b  = ft * 16 + l15;  // B-matrix N index (f)

  // per-lane contiguous A row: adjsT[k][j_a][*]
  const float* __restrict__ arow = adjsT + (unsigned)(k * E_ + j_a) * E_;

  v8f cp = {};  // Σ adj_agg · Hp
  v8f cm = {};  // Σ adj_agg · Hm
#pragma unroll 4
  for (int i0 = 0; i0 < E_; i0 += 4) {
    const int ia = i0 + kb;                       // even -> 8B aligned
    const v2f a = *reinterpret_cast<const v2f*>(arow + ia);   // global_load_b64
    // B[k=i_local, n=f]: Hp / Hm split of h (lane-coalesced: one line / 16 lanes)
    const float h0 = h[(ia    ) * F_ + f_b];
    const float h1 = h[(ia + 1) * F_ + f_b];
    v2f bp, bm;
    bp.x = fmaxf(h0, 0.f); bp.y = fmaxf(h1, 0.f);
    bm.x = fminf(h0, 0.f); bm.y = fminf(h1, 0.f);
    cp = __builtin_amdgcn_wmma_f32_16x16x4_f32(false, a, false, bp, (short)0, cp, false, false);
    cm = __builtin_amdgcn_wmma_f32_16x16x4_f32(false, a, false, bm, (short)0, cm, false, false);
  }
  // C/D layout: VGPR v -> row M = v + half*8, col N = l15
#pragma unroll
  for (int v = 0; v < 8; ++v) {
    const int j = jt * 16 + v + half * 8;
    const int f = ft * 16 + l15;
    const float s  = h[j * F_ + f] * r[k * F_ + f];
    const float ap = cp[v], am = cm[v];
    const float a_att = (s > 0.f) ? (s * (ap + LRELU_ALPHA * am))
                                  : (s * (LRELU_ALPHA * ap + am));
    att[(j * R_ + k) * F_ + f] = a_att;
  }
}

// K3: per (k,f): max/sumexp over e (length 256). grid = R*F, block = 256 (e)
__global__ void k_colstats(const float* __restrict__ att,
                           float* __restrict__ colmax, float* __restrict__ colsum) {
  __shared__ float red[256];
  const int b = blockIdx.x;              // k*64+f
  const int k = b >> 6, f = b & 63;
  const int t = threadIdx.x;             // e
  const float v = att[(t * R_ + k) * F_ + f];
  red[t] = v; __syncthreads();
  for (int s = 128; s > 0; s >>= 1) { if (t < s) red[t] = fmaxf(red[t], red[t + s]); __syncthreads(); }
  const float m = red[0]; __syncthreads();
  red[t] = __expf(v - m); __syncthreads();
  for (int s = 128; s > 0; s >>= 1) { if (t < s) red[t] += red[t + s]; __syncthreads(); }
  if (t == 0) { colmax[b] = m; colsum[b] = red[0]; }
}

// K4: per (e,f): max/sumexp over k (length 16). 16384 threads.
__global__ void k_rowstats(const float* __restrict__ att,
                           float* __restrict__ rowmax, float* __restrict__ rowsum) {
  const int tid = blockIdx.x * blockDim.x + threadIdx.x;  // e*64+f
  const int e = tid >> 6, f = tid & 63;
  float m = -3.402823466e38f;
  for (int k = 0; k < R_; ++k) m = fmaxf(m, att[(e * R_ + k) * F_ + f]);
  float s = 0.f;
  for (int k = 0; k < R_; ++k) s += __expf(att[(e * R_ + k) * F_ + f] - m);
  rowmax[tid] = m; rowsum[tid] = s;
}

// K5: h' = elu(h .* Σ_k attention_E). 16384 threads.
__global__ void k_hprime(const float* __restrict__ att, const float* __restrict__ h,
                         const float* __restrict__ colmax, const float* __restrict__ colsum,
                         float* __restrict__ out) {
  const int tid = blockIdx.x * blockDim.x + threadIdx.x;  // e*64+f
  const int e = tid >> 6, f = tid & 63;
  float acc = 0.f;
  for (int k = 0; k < R_; ++k)
    acc += __expf(att[(e * R_ + k) * F_ + f] - colmax[k * F_ + f]) / colsum[k * F_ + f];
  const float hp = h[tid] * acc;
  out[tid] = hp > 0.f ? hp : expm1f(hp);
}

// K6: r' = elu(r .* Σ_e attention_R). grid = R*F, block = 256 (e)
__global__ void k_rprime(const float* __restrict__ att, const float* __restrict__ r,
                         const float* __restrict__ rowmax, const float* __restrict__ rowsum,
                         float* __restrict__ out) {
  __shared__ float red[256];
  const int b = blockIdx.x;              // k*64+f
  const int k = b >> 6, f = b & 63;
  const int t = threadIdx.x;             // e
  red[t] = __expf(att[(t * R_ + k) * F_ + f] - rowmax[t * F_ + f]) / rowsum[t * F_ + f];
  __syncthreads();
  for (int s = 128; s > 0; s >>= 1) { if (t < s) red[t] += red[t + s]; __syncthreads(); }
  if (t == 0) {
    const float rp = r[k * F_ + f] * red[0];
    out[E_ * F_ + b] = rp > 0.f ? rp : expm1f(rp);
  }
}

// K7: alpha logits[e,k] = Σ_f attention_R[e,k,f]*w[f] + b. 4096 threads.
__global__ void k_alpha_logits(const float* __restrict__ att,
                               const float* __restrict__ rowmax, const float* __restrict__ rowsum,
                               const float* __restrict__ w, const float* __restrict__ bb,
                               float* __restrict__ logits) {
  const int tid = blockIdx.x * blockDim.x + threadIdx.x;  // e*16+k
  const int e = tid >> 4, k = tid & 15;
  float acc = 0.f;
  for (int f = 0; f < F_; ++f)
    acc += (__expf(att[(e * R_ + k) * F_ + f] - rowmax[e * F_ + f]) / rowsum[e * F_ + f]) * w[f];
  logits[tid] = acc + bb[0];
}

// K8: alpha = softmax over 256-element groups of flat (e*16+k) order. grid = 16, block = 256.
__global__ void k_alpha_softmax(const float* __restrict__ logits, float* __restrict__ out) {
  __shared__ float red[256];
  const int p = blockIdx.x, t = threadIdx.x;
  const float v = logits[p * 256 + t];
  red[t] = v; __syncthreads();
  for (int s = 128; s > 0; s >>= 1) { if (t < s) red[t] = fmaxf(red[t], red[t + s]); __syncthreads(); }
  const float m = red[0]; __syncthreads();
  const float e = __expf(v - m);
  red[t] = e; __syncthreads();
  for (int s = 128; s > 0; s >>= 1) { if (t < s) red[t] += red[t + s]; __syncthreads(); }
  out[E_ * F_ + R_ * F_ + p * 256 + t] = e / red[0];
}

extern "C" void kernel_launch(void* const* d_in, const int* in_sizes, int n_in,
                              void* d_out, int out_size, void* d_ws, size_t ws_size,
                              hipStream_t stream) {
  const float* h     = (const float*)d_in[0];   // (256,64)
  const float* r     = (const float*)d_in[1];   // (16,64)
  const float* adj   = (const float*)d_in[2];   // (256,256,16)
  const float* w_lin = (const float*)d_in[3];   // (1,64)
  const float* b_lin = (const float*)d_in[4];   // (1,)
  float* out = (float*)d_out;                   // 16384 + 1024 + 4096

  float* ws     = (float*)d_ws;
  float* adjsT  = ws + OFF_ADJS;
  float* att    = ws + OFF_ATT;
  float* colmax = ws + OFF_CMAX;
  float* colsum = ws + OFF_CSUM;
  float* rowmax = ws + OFF_RMAX;
  float* rowsum = ws + OFF_RSUM;
  float* logits = ws + OFF_LOG;

  k_softmax_adj   <<<E_ * R_, 256, 0, stream>>>(adj, adjsT);
  k_attention_wmma<<<16 * 16 * 4, 32, 0, stream>>>(adjsT, h, r, att);
  k_colstats      <<<R_ * F_, 256, 0, stream>>>(att, colmax, colsum);
  k_rowstats      <<<(E_ * F_) / 256, 256, 0, stream>>>(att, rowmax, rowsum);
  k_hprime        <<<(E_ * F_) / 256, 256, 0, stream>>>(att, h, colmax, colsum, out);
  k_rprime        <<<R_ * F_, 256, 0, stream>>>(att, r, rowmax, rowsum, out);
  k_alpha_logits  <<<(E_ * R_) / 256, 256, 0, stream>>>(att, rowmax, rowsum, w_lin, b_lin, logits);
  k_alpha_softmax <<<16, 256, 0, stream>>>(logits, out);
}